// GCN_40080634806795
// MI455X (gfx1250) — compile-verified
//
#include <hip/hip_runtime.h>
#include <hip/hip_bf16.h>
#include <math.h>

// ---------------------------------------------------------------------------
// GCN forward for MI455X (gfx1250, wave32).
//   deg -> dinv -> 3 x [ WMMA-f32 GEMM ; self-loop init ; edge scatter ;
//                        bias+relu + pooled accumulation ] -> head+softmax
// GEMMs use V_WMMA_F32_16X16X4_F32 (full fp32 matrix pipe).
// ---------------------------------------------------------------------------

typedef float v2f __attribute__((ext_vector_type(2)));
typedef float v8f __attribute__((ext_vector_type(8)));

#define F_SIZE   64
#define N_GRAPHS 256
#define OUT_DIM  10

// ---------------------------------------------------------------- zero range
__global__ void gcn_zero_kernel(float* __restrict__ p, int n) {
    int i = blockIdx.x * blockDim.x + threadIdx.x;
    if (i < n) p[i] = 0.0f;
}

// ------------------------------------------------------------------- degrees
// deg[dst] += |w| over edges (self-loop +1 folded into rsqrt below)
__global__ void gcn_degree_kernel(const int* __restrict__ ei,
                                  const float* __restrict__ w,
                                  float* __restrict__ deg, int nE) {
    int e = blockIdx.x * blockDim.x + threadIdx.x;
    if (e >= nE) return;
    atomicAdd(&deg[ei[nE + e]], fabsf(w[e]));
}

// dinv = rsqrt(deg + 1);  also count nodes per graph
__global__ void gcn_dinv_counts_kernel(float* __restrict__ deg,
                                       const int* __restrict__ batch,
                                       float* __restrict__ counts, int n) {
    int i = blockIdx.x * blockDim.x + threadIdx.x;
    if (i >= n) return;
    deg[i] = rsqrtf(deg[i] + 1.0f);
    atomicAdd(&counts[batch[i]], 1.0f);
}

// ------------------------------------------------------------ WMMA f32 GEMM
// H[n, 64] = A[n, K] @ W[K, 64]   (one wave per 16x16 output tile)
// A-frag (16x4 f32): lanes 0-15 hold K = k+0..k+1, lanes 16-31 hold K = k+2..k+3
// B-frag (4x16 f32): VGPR0 = row k (+2 for hi half-wave), VGPR1 = row k+1 (+3)
// D (16x16 f32): vgpr i -> M = i + 8*(lane>=16), N = lane&15
__global__ void gcn_gemm_wmma_kernel(const float* __restrict__ A,
                                     const float* __restrict__ W,
                                     float* __restrict__ H,
                                     int n, int K) {
    int wave = (blockIdx.x * blockDim.x + threadIdx.x) >> 5;
    int lane = threadIdx.x & 31;
    int rowTile = wave >> 2;            // 4 column tiles of 16 cover F_SIZE=64
    int colTile = wave & 3;
    if (rowTile * 16 >= n) return;      // uniform per wave
    int m0 = rowTile * 16;
    int n0 = colTile * 16;
    int lhalf = lane >> 4;              // 0: K+0..1   1: K+2..3
    int l15   = lane & 15;

    const float* arow = A + (size_t)(m0 + l15) * K + 2 * lhalf;
    const float* wcol = W + (size_t)(2 * lhalf) * F_SIZE + n0 + l15;

    v8f acc = {};
    for (int k = 0; k < K; k += 4) {
        v2f a = *(const v2f*)(arow + k);
        v2f b;
        b.x = wcol[(size_t)k * F_SIZE];
        b.y = wcol[(size_t)(k + 1) * F_SIZE];
        acc = __builtin_amdgcn_wmma_f32_16x16x4_f32(
            /*neg_a=*/false, a, /*neg_b=*/false, b,
            /*c_mod=*/(short)0, acc, /*reuse_a=*/false, /*reuse_b=*/false);
    }

    float* hrow = H + (size_t)(m0 + 8 * lhalf) * F_SIZE + n0 + l15;
#pragma unroll
    for (int i = 0; i < 8; ++i)
        hrow[(size_t)i * F_SIZE] = acc[i];
}

// ------------------------------------------------- self-loop: out = dinv^2*h
__global__ void gcn_selfloop_kernel(const float* __restrict__ h,
                                    const float* __restrict__ dinv,
                                    float* __restrict__ out, int total) {
    int i = blockIdx.x * blockDim.x + threadIdx.x;
    if (i >= total) return;
    float d = dinv[i >> 6];
    out[i] = d * d * h[i];
}

// --------------------------------------------- edge scatter (16 threads/edge)
__global__ void gcn_scatter_kernel(const int* __restrict__ ei,
                                   const float* __restrict__ w,
                                   const float* __restrict__ dinv,
                                   const float* __restrict__ h,
                                   float* __restrict__ out, int nE) {
    int t = blockIdx.x * blockDim.x + threadIdx.x;
    int e  = t >> 4;
    int f4 = (t & 15) << 2;             // float4 chunk within the 64-wide row
    if (e >= nE) return;
    int src = ei[e];
    int dst = ei[nE + e];
    float c = dinv[src] * fabsf(w[e]) * dinv[dst];
    const float4 hv = *(const float4*)(h + (size_t)src * F_SIZE + f4);
    float* o = out + (size_t)dst * F_SIZE + f4;
    atomicAdd(o + 0, c * hv.x);
    atomicAdd(o + 1, c * hv.y);
    atomicAdd(o + 2, c * hv.z);
    atomicAdd(o + 3, c * hv.w);
}

// ------------------- bias + relu in place; accumulate pooled per-graph sums
__global__ void gcn_finalize_kernel(float* __restrict__ out,
                                    const float* __restrict__ bias,
                                    const int* __restrict__ batch,
                                    float* __restrict__ pooled, int total) {
    int i = blockIdx.x * blockDim.x + threadIdx.x;
    if (i >= total) return;
    int node = i >> 6;
    int f    = i & 63;
    float v = out[i] + bias[f];
    v = v > 0.0f ? v : 0.0f;
    out[i] = v;
    atomicAdd(&pooled[(size_t)batch[node] * F_SIZE + f], v);
}

// ------------------------------------- head: mean -> 64x10 GEMV -> softmax
__global__ void gcn_head_kernel(const float* __restrict__ pooled,
                                const float* __restrict__ counts,
                                const float* __restrict__ Wl,
                                const float* __restrict__ bl,
                                float* __restrict__ outp) {
    int g = blockIdx.x * blockDim.x + threadIdx.x;
    if (g >= N_GRAPHS) return;
    float inv = 1.0f / (3.0f * fmaxf(counts[g], 1.0f));   // /3 layer avg, /cnt mean
    float logits[OUT_DIM];
#pragma unroll
    for (int o = 0; o < OUT_DIM; ++o) logits[o] = bl[o];
    for (int f = 0; f < F_SIZE; ++f) {
        float p = pooled[(size_t)g * F_SIZE + f] * inv;
#pragma unroll
        for (int o = 0; o < OUT_DIM; ++o)
            logits[o] += p * Wl[f * OUT_DIM + o];
    }
    float m = logits[0];
#pragma unroll
    for (int o = 1; o < OUT_DIM; ++o) m = fmaxf(m, logits[o]);
    float s = 0.0f;
#pragma unroll
    for (int o = 0; o < OUT_DIM; ++o) { logits[o] = expf(logits[o] - m); s += logits[o]; }
    float rs = 1.0f / s;
#pragma unroll
    for (int o = 0; o < OUT_DIM; ++o) outp[(size_t)g * OUT_DIM + o] = logits[o] * rs;
}

// ---------------------------------------------------------------------------
extern "C" void kernel_launch(void* const* d_in, const int* in_sizes, int n_in,
                              void* d_out, int out_size, void* d_ws, size_t ws_size,
                              hipStream_t stream) {
    const float* x    = (const float*)d_in[0];   // [n, 128]
    const float* ew   = (const float*)d_in[1];   // [nE]
    const float* W1   = (const float*)d_in[2];
    const float* b1   = (const float*)d_in[3];
    const float* W2   = (const float*)d_in[4];
    const float* b2   = (const float*)d_in[5];
    const float* W3   = (const float*)d_in[6];
    const float* b3   = (const float*)d_in[7];
    const float* Wl   = (const float*)d_in[8];
    const float* bl   = (const float*)d_in[9];
    const int*   ei   = (const int*)d_in[10];    // [2, nE] flat
    const int*   bat  = (const int*)d_in[11];    // [n]

    const int nE = in_sizes[1];
    const int n  = in_sizes[11];
    const int IN_DIM = in_sizes[0] / n;          // 128

    // ----- workspace layout (floats) -----
    float* ws      = (float*)d_ws;
    float* deg     = ws;                              // n      (becomes dinv)
    float* pooled  = deg + n;                         // 256*64
    float* counts  = pooled + N_GRAPHS * F_SIZE;      // 256
    float* h       = counts + N_GRAPHS;               // n*64
    float* bufA    = h    + (size_t)n * F_SIZE;       // n*64
    float* bufB    = bufA + (size_t)n * F_SIZE;       // n*64

    const int TB = 256;
    const int zeroN = n + N_GRAPHS * F_SIZE + N_GRAPHS;
    const int totalF = n * F_SIZE;

    // zero deg + pooled + counts (contiguous)
    gcn_zero_kernel<<<(zeroN + TB - 1) / TB, TB, 0, stream>>>(deg, zeroN);

    // degrees and dinv / counts
    gcn_degree_kernel<<<(nE + TB - 1) / TB, TB, 0, stream>>>(ei, ew, deg, nE);
    gcn_dinv_counts_kernel<<<(n + TB - 1) / TB, TB, 0, stream>>>(deg, bat, counts, n);

    const int rowTiles   = (n + 15) / 16;
    const int gemmWaves  = rowTiles * (F_SIZE / 16);
    const int gemmBlocks = (gemmWaves * 32 + TB - 1) / TB;
    const int elemBlocks = (totalF + TB - 1) / TB;
    const int scatBlocks = ((size_t)nE * 16 + TB - 1) / TB;

    // ---- layer 1: x(128) -> bufA ----
    gcn_gemm_wmma_kernel<<<gemmBlocks, TB, 0, stream>>>(x, W1, h, n, IN_DIM);
    gcn_selfloop_kernel<<<elemBlocks, TB, 0, stream>>>(h, deg, bufA, totalF);
    gcn_scatter_kernel<<<scatBlocks, TB, 0, stream>>>(ei, ew, deg, h, bufA, nE);
    gcn_finalize_kernel<<<elemBlocks, TB, 0, stream>>>(bufA, b1, bat, pooled, totalF);

    // ---- layer 2: bufA -> bufB ----
    gcn_gemm_wmma_kernel<<<gemmBlocks, TB, 0, stream>>>(bufA, W2, h, n, F_SIZE);
    gcn_selfloop_kernel<<<elemBlocks, TB, 0, stream>>>(h, deg, bufB, totalF);
    gcn_scatter_kernel<<<scatBlocks, TB, 0, stream>>>(ei, ew, deg, h, bufB, nE);
    gcn_finalize_kernel<<<elemBlocks, TB, 0, stream>>>(bufB, b2, bat, pooled, totalF);

    // ---- layer 3: bufB -> bufA (x1 no longer needed) ----
    gcn_gemm_wmma_kernel<<<gemmBlocks, TB, 0, stream>>>(bufB, W3, h, n, F_SIZE);
    gcn_selfloop_kernel<<<elemBlocks, TB, 0, stream>>>(h, deg, bufA, totalF);
    gcn_scatter_kernel<<<scatBlocks, TB, 0, stream>>>(ei, ew, deg, h, bufA, nE);
    gcn_finalize_kernel<<<elemBlocks, TB, 0, stream>>>(bufA, b3, bat, pooled, totalF);

    // ---- head ----
    gcn_head_kernel<<<1, N_GRAPHS, 0, stream>>>(pooled, counts, Wl, bl, (float*)d_out);
}